// Aggregator_69200513073772
// MI455X (gfx1250) — compile-verified
//
#include <hip/hip_runtime.h>

// Problem constants from the reference: B=1024, E=64, N=32, D=64.
#define DIM        64
#define NNBR       32
#define E_PER_B    64
#define ROWS_TOTAL (1024 * 64)       // B*E flattened rows
#define ROWS_PER_WG 128
#define WG_SIZE     256              // 8 waves (wave32)
#define NUM_WG     (ROWS_TOTAL / ROWS_PER_WG)   // 512
#define LDS_STRIDE  68               // 64 + 4 pad -> conflict-free WMMA frag reads

typedef __attribute__((ext_vector_type(2))) float v2f;
typedef __attribute__((ext_vector_type(8))) float v8f;

__global__ __launch_bounds__(WG_SIZE)
void Aggregator_69200513073772_kernel(const float* __restrict__ self_vectors,
                                      const float* __restrict__ neighbor_vectors,
                                      const float* __restrict__ neighbor_relations,
                                      const float* __restrict__ user_embeddings,
                                      const float* __restrict__ Wm,
                                      const float* __restrict__ bias,
                                      float* __restrict__ out)
{
    __shared__ float Xs[ROWS_PER_WG][LDS_STRIDE];  // X = self + agg (per-WG row tile)
    __shared__ float Ws[DIM][LDS_STRIDE];          // W staged row-major
    __shared__ float Us[2][DIM];                   // user embeddings for the 2 b's this WG spans

    const int tid       = threadIdx.x;
    const int lane      = tid & 31;
    const int wave      = tid >> 5;
    const int wgRowBase = blockIdx.x * ROWS_PER_WG;

    // ---- stage W and u into LDS ----
    for (int idx = tid; idx < DIM * DIM; idx += WG_SIZE)
        Ws[idx >> 6][idx & 63] = Wm[idx];
    if (tid < 2 * DIM) {
        const int bb = wgRowBase / E_PER_B + (tid >> 6);
        Us[tid >> 6][tid & 63] = user_embeddings[bb * DIM + (tid & 63)];
    }
    __syncthreads();

    // ================= Phase 1: attention + aggregation (HBM streaming) =========
    // Each wave owns 16 consecutive (b,e) rows; one row per iteration.
    for (int i = 0; i < 16; ++i) {
        const int  localRow = wave * 16 + i;
        const long row      = (long)(wgRowBase + localRow);
        const float* relRow  = neighbor_relations + row * (NNBR * DIM) + lane * DIM;
        const float* vecBase = neighbor_vectors  + row * (NNBR * DIM);

        // Warm the V stream (global_prefetch_b8) while we chew on relations.
        __builtin_prefetch(vecBase + lane * DIM, 0, 1);

        // score[n=lane] = <u[b], rel[row, lane, :]>   (b128 loads, full row per lane)
        const float4* r4p = (const float4*)relRow;
        const float4* u4p = (const float4*)(&Us[localRow >> 6][0]); // LDS broadcast
        float s = 0.f;
#pragma unroll
        for (int k = 0; k < DIM / 4; ++k) {
            const float4 r4 = r4p[k];
            const float4 u4 = u4p[k];
            s = fmaf(r4.x, u4.x, s);
            s = fmaf(r4.y, u4.y, s);
            s = fmaf(r4.z, u4.z, s);
            s = fmaf(r4.w, u4.w, s);
        }

        // self-masked softmax over the 32 neighbors (wave32 butterfly reduce)
        float ee  = (s != 0.f) ? __expf(s) : 0.f;
        float tot = ee;
#pragma unroll
        for (int off = 16; off >= 1; off >>= 1)
            tot += __shfl_xor(tot, off, 32);
        tot = (tot != 0.f) ? tot : 1.f;
        const float attn = ee / tot;

        // agg[d] = sum_n attn[n] * V[n,d]; lane l owns d = {2l, 2l+1} so every
        // V-row load is one fully-coalesced 256B transaction across the wave.
        float2 acc = make_float2(0.f, 0.f);
#pragma unroll
        for (int n = 0; n < NNBR; ++n) {
            const float  a = __shfl(attn, n, 32);
            const float2 v = *(const float2*)(vecBase + n * DIM + 2 * lane);
            acc.x = fmaf(a, v.x, acc.x);
            acc.y = fmaf(a, v.y, acc.y);
        }

        const float2 sv = *(const float2*)(self_vectors + row * DIM + 2 * lane);
        Xs[localRow][2 * lane]     = acc.x + sv.x;
        Xs[localRow][2 * lane + 1] = acc.y + sv.y;
    }
    __syncthreads();

    // ================= Phase 2: out = relu(X @ W^T + b) via V_WMMA_F32_16X16X4_F32
    // Each wave computes a 16x64 output tile: 4 N-tiles x 16 K-steps = 64 WMMAs.
    // A frag (16x4 f32): lane<16 holds {X[m=nl][k0], X[nl][k0+1]},
    //                    lane>=16 holds {X[nl][k0+2], X[nl][k0+3]}.
    // B frag (4x16, B[k][n] = W[n][k]) mirrors the A striping.
    const int rowBase = wave * 16;
    const int nl      = lane & 15;
    const int hs      = (lane < 16) ? 0 : 2;

    v8f acc0, acc1, acc2, acc3;
    {
        const float b0 = bias[ 0 + nl], b1 = bias[16 + nl];
        const float b2 = bias[32 + nl], b3 = bias[48 + nl];
        acc0 = (v8f){b0, b0, b0, b0, b0, b0, b0, b0};
        acc1 = (v8f){b1, b1, b1, b1, b1, b1, b1, b1};
        acc2 = (v8f){b2, b2, b2, b2, b2, b2, b2, b2};
        acc3 = (v8f){b3, b3, b3, b3, b3, b3, b3, b3};
    }

#pragma unroll
    for (int k0 = 0; k0 < DIM; k0 += 4) {
        const float2 a2 = *(const float2*)(&Xs[rowBase + nl][k0 + hs]);
        v2f A; A.x = a2.x; A.y = a2.y;

        const float2 w0 = *(const float2*)(&Ws[ 0 + nl][k0 + hs]);
        const float2 w1 = *(const float2*)(&Ws[16 + nl][k0 + hs]);
        const float2 w2 = *(const float2*)(&Ws[32 + nl][k0 + hs]);
        const float2 w3 = *(const float2*)(&Ws[48 + nl][k0 + hs]);
        v2f B0; B0.x = w0.x; B0.y = w0.y;
        v2f B1; B1.x = w1.x; B1.y = w1.y;
        v2f B2; B2.x = w2.x; B2.y = w2.y;
        v2f B3; B3.x = w3.x; B3.y = w3.y;

        acc0 = __builtin_amdgcn_wmma_f32_16x16x4_f32(false, A, false, B0, (short)0, acc0, false, false);
        acc1 = __builtin_amdgcn_wmma_f32_16x16x4_f32(false, A, false, B1, (short)0, acc1, false, false);
        acc2 = __builtin_amdgcn_wmma_f32_16x16x4_f32(false, A, false, B2, (short)0, acc2, false, false);
        acc3 = __builtin_amdgcn_wmma_f32_16x16x4_f32(false, A, false, B3, (short)0, acc3, false, false);
    }

    // C/D layout: VGPR r -> lanes 0-15: C[r][nl], lanes 16-31: C[r+8][nl]
    float* outp = out + (long)(wgRowBase + rowBase) * DIM;
    const int mOff = (lane < 16) ? 0 : 8;
#pragma unroll
    for (int r = 0; r < 8; ++r) {
        const long m = (long)(r + mOff);
        outp[m * DIM +  0 + nl] = fmaxf(acc0[r], 0.f);
        outp[m * DIM + 16 + nl] = fmaxf(acc1[r], 0.f);
        outp[m * DIM + 32 + nl] = fmaxf(acc2[r], 0.f);
        outp[m * DIM + 48 + nl] = fmaxf(acc3[r], 0.f);
    }
}

extern "C" void kernel_launch(void* const* d_in, const int* in_sizes, int n_in,
                              void* d_out, int out_size, void* d_ws, size_t ws_size,
                              hipStream_t stream) {
    const float* self_vectors       = (const float*)d_in[0]; // (B,E,1,D) f32
    const float* neighbor_vectors   = (const float*)d_in[1]; // (B,E,N,D) f32
    const float* neighbor_relations = (const float*)d_in[2]; // (B,E,N,D) f32
    // d_in[3] = masks (bool) -- unused by the reference computation
    const float* user_embeddings    = (const float*)d_in[4]; // (B,D) f32
    const float* Wm                 = (const float*)d_in[5]; // (D,D) f32
    const float* bias               = (const float*)d_in[6]; // (D,) f32
    float* out = (float*)d_out;                              // (B,E,D) f32

    (void)in_sizes; (void)n_in; (void)out_size; (void)d_ws; (void)ws_size;

    hipLaunchKernelGGL(Aggregator_69200513073772_kernel,
                       dim3(NUM_WG), dim3(WG_SIZE), 0, stream,
                       self_vectors, neighbor_vectors, neighbor_relations,
                       user_embeddings, Wm, bias, out);
}